// ToxGCN_18614388261451
// MI455X (gfx1250) — compile-verified
//
#include <hip/hip_runtime.h>

// ToxGCN on gfx1250: 3x (GEMM -> edge scatter (D^-1/2 A D^-1/2) -> BN+ReLU),
// graph mean pool, MLP head. fp32 WMMA (V_WMMA_F32_16X16X4_F32) for the
// 128x128 GEMMs; hardware global_atomic_add_f32 for the scatter.

#define N_NODES   1000000
#define N_EDGES   4000000
#define N_GRAPHS  32768
#define IN_DIM    7
#define HID       128
#define BN_EPS    1e-5f

typedef __attribute__((ext_vector_type(2))) float v2f;
typedef __attribute__((ext_vector_type(8))) float v8f;

__device__ __forceinline__ void atomAddF(float* p, float v) {
  // hardware global_atomic_add_f32 (no CAS loop)
  unsafeAtomicAdd(p, v);
}

// ---------------------------------------------------------------- degrees ---
__global__ void deg_kernel(const int* __restrict__ dst, float* __restrict__ deg, int E) {
  int e = blockIdx.x * blockDim.x + threadIdx.x;
  if (e < E) atomAddF(&deg[dst[e]], 1.0f);
}

__global__ void dinv_kernel(float* __restrict__ deg, int N) {
  int i = blockIdx.x * blockDim.x + threadIdx.x;
  if (i < N) deg[i] = __frsqrt_rn(deg[i] + 1.0f);  // +1 for the self loop
}

// ------------------------------------------------------- layer-1 GEMM (K=7) -
// one wave per node; lane handles 4 output channels
__global__ __launch_bounds__(256) void gemm_in7(const float* __restrict__ x,
                                                const float* __restrict__ W,
                                                float* __restrict__ T, int N) {
  long t = (long)blockIdx.x * blockDim.x + threadIdx.x;
  long node = t >> 5;
  int lane = (int)(t & 31);
  if (node >= N) return;
  float xv[IN_DIM];
#pragma unroll
  for (int k = 0; k < IN_DIM; ++k) xv[k] = x[node * IN_DIM + k];
  int c0 = lane * 4;
  float4 acc = {0.f, 0.f, 0.f, 0.f};
#pragma unroll
  for (int k = 0; k < IN_DIM; ++k) {
    const float* wr = W + k * HID + c0;
    acc.x = fmaf(xv[k], wr[0], acc.x);
    acc.y = fmaf(xv[k], wr[1], acc.y);
    acc.z = fmaf(xv[k], wr[2], acc.z);
    acc.w = fmaf(xv[k], wr[3], acc.w);
  }
  ((float4*)(T + node * HID))[lane] = acc;
}

// ---------------------------------------------- 128x128 GEMM via fp32 WMMA --
// block = 256 threads = 8 waves, each wave computes a 16x128 output strip.
// W (128x128, row-major fi x fo) is staged in LDS once per block.
__global__ __launch_bounds__(256) void gemm128_wmma(const float* __restrict__ A,
                                                    const float* __restrict__ W,
                                                    float* __restrict__ B, int N) {
  __shared__ float ldsW[HID * HID];  // 64 KB
  {
    const float4* Wv = (const float4*)W;
    float4* Lv = (float4*)ldsW;
    for (int i = threadIdx.x; i < HID * HID / 4; i += 256) Lv[i] = Wv[i];
  }
  __syncthreads();

  const int wave = threadIdx.x >> 5;
  const int lane = threadIdx.x & 31;
  const int m    = lane & 15;   // M (rows) / N (cols) index within tile
  const int half = lane >> 4;   // K-pair selector per ISA A/B layout

  long r0   = (long)blockIdx.x * 128 + wave * 16;
  long rowA = r0 + m;
  if (rowA >= N) rowA = N - 1;  // clamp loads; stores are guarded (EXEC all-1)
  const v2f* arow = (const v2f*)(A + rowA * HID);

  // A strip in registers: lane holds A[M=m][4k+2*half .. +1] for k=0..31
  v2f afrag[32];
#pragma unroll
  for (int k = 0; k < 32; ++k) afrag[k] = arow[2 * k + half];

#pragma unroll
  for (int nc = 0; nc < 8; ++nc) {
    const int col = nc * 16 + m;
    v8f acc = {0.f, 0.f, 0.f, 0.f, 0.f, 0.f, 0.f, 0.f};
#pragma unroll
    for (int k = 0; k < 32; ++k) {
      const int kr = 4 * k + 2 * half;
      v2f b;
      b.x = ldsW[kr * HID + col];
      b.y = ldsW[(kr + 1) * HID + col];
      acc = __builtin_amdgcn_wmma_f32_16x16x4_f32(
          false, afrag[k], false, b, (short)0, acc, false, false);
    }
    // C/D layout: VGPR i -> M = i + 8*half, N = m
#pragma unroll
    for (int i = 0; i < 8; ++i) {
      long ro = r0 + half * 8 + i;
      if (ro < N) B[ro * HID + col] = acc[i];
    }
  }
}

// ---------------------------------------------------------- edge scatter ----
// one wave per edge; lane handles float4 of the 128 channels
__global__ __launch_bounds__(256) void scatter_edges(const int* __restrict__ src,
                                                     const int* __restrict__ dst,
                                                     const float* __restrict__ dinv,
                                                     const float* __restrict__ T,
                                                     float* __restrict__ Agg, int E) {
  long t = (long)blockIdx.x * blockDim.x + threadIdx.x;
  long e = t >> 5;
  int lane = (int)(t & 31);
  if (e >= E) return;
  int s = src[e], d = dst[e];
  float w = dinv[s] * dinv[d];
  float4 v = ((const float4*)(T + (long)s * HID))[lane];
  float* o = Agg + (long)d * HID + lane * 4;
  atomAddF(o + 0, v.x * w);
  atomAddF(o + 1, v.y * w);
  atomAddF(o + 2, v.z * w);
  atomAddF(o + 3, v.w * w);
}

// -------------------------------------- self-loop + bias + BN + ReLU fuse ---
__global__ __launch_bounds__(256) void epilogue(const float* __restrict__ Agg,
                                                const float* __restrict__ T,
                                                const float* __restrict__ dinv,
                                                const float* __restrict__ bias,
                                                const float* __restrict__ gamma,
                                                const float* __restrict__ beta,
                                                const float* __restrict__ rmean,
                                                const float* __restrict__ rvar,
                                                float* __restrict__ H, int N) {
  long t = (long)blockIdx.x * blockDim.x + threadIdx.x;
  long node = t >> 5;
  int lane = (int)(t & 31);
  if (node >= N) return;
  float di = dinv[node];
  float sl = di * di;  // self-loop norm
  int c0 = lane * 4;
  float4 a  = ((const float4*)(Agg + node * HID))[lane];
  float4 tv = ((const float4*)(T + node * HID))[lane];
  float va[4] = {a.x, a.y, a.z, a.w};
  float vt[4] = {tv.x, tv.y, tv.z, tv.w};
  float vo[4];
#pragma unroll
  for (int j = 0; j < 4; ++j) {
    int c = c0 + j;
    float s = gamma[c] * __frsqrt_rn(rvar[c] + BN_EPS);
    float val = va[j] + sl * vt[j] + bias[c];
    val = (val - rmean[c]) * s + beta[c];
    vo[j] = fmaxf(val, 0.0f);
  }
  float4 o = {vo[0], vo[1], vo[2], vo[3]};
  ((float4*)(H + node * HID))[lane] = o;
}

// ------------------------------------------------------------- pooling ------
__global__ void graph_count(const int* __restrict__ batch, float* __restrict__ cnt, int N) {
  int i = blockIdx.x * blockDim.x + threadIdx.x;
  if (i < N) atomAddF(&cnt[batch[i]], 1.0f);
}

__global__ __launch_bounds__(256) void pool_sum(const float* __restrict__ H,
                                                const int* __restrict__ batch,
                                                float* __restrict__ pooled, int N) {
  long t = (long)blockIdx.x * blockDim.x + threadIdx.x;
  long node = t >> 5;
  int lane = (int)(t & 31);
  if (node >= N) return;
  int g = batch[node];
  float4 v = ((const float4*)(H + node * HID))[lane];
  float* o = pooled + (long)g * HID + lane * 4;
  atomAddF(o + 0, v.x);
  atomAddF(o + 1, v.y);
  atomAddF(o + 2, v.z);
  atomAddF(o + 3, v.w);
}

// ------------------------------------------------------------ MLP head ------
// one wave per graph; lane owns 2 hidden units, wave-reduce for the logit
__global__ __launch_bounds__(256) void mlp_head(const float* __restrict__ pooled,
                                                const float* __restrict__ cnt,
                                                const float* __restrict__ Wc1,
                                                const float* __restrict__ bc1,
                                                const float* __restrict__ Wc2,
                                                const float* __restrict__ bc2,
                                                float* __restrict__ out, int G) {
  long t = (long)blockIdx.x * blockDim.x + threadIdx.x;
  long g = t >> 5;
  int lane = (int)(t & 31);
  if (g >= G) return;
  float inv = 1.0f / fmaxf(cnt[g], 1.0f);
  int u0 = lane * 2, u1 = lane * 2 + 1;
  float h0 = bc1[u0], h1 = bc1[u1];
  const float* pr = pooled + g * HID;
#pragma unroll 4
  for (int k = 0; k < HID; ++k) {
    float p = pr[k] * inv;
    h0 = fmaf(p, Wc1[k * (HID / 2) + u0], h0);
    h1 = fmaf(p, Wc1[k * (HID / 2) + u1], h1);
  }
  h0 = fmaxf(h0, 0.0f);
  h1 = fmaxf(h1, 0.0f);
  float part = h0 * Wc2[u0] + h1 * Wc2[u1];
#pragma unroll
  for (int off = 16; off > 0; off >>= 1) part += __shfl_down(part, off, 32);
  if (lane == 0) out[g] = part + bc2[0];
}

// ---------------------------------------------------------------------------
extern "C" void kernel_launch(void* const* d_in, const int* in_sizes, int n_in,
                              void* d_out, int out_size, void* d_ws, size_t ws_size,
                              hipStream_t stream) {
  const float* x      = (const float*)d_in[0];
  const int*   eidx   = (const int*)d_in[1];
  const int*   batch  = (const int*)d_in[2];
  const int*   src    = eidx;
  const int*   dst    = eidx + N_EDGES;

  const float* W[3]     = {(const float*)d_in[3],  (const float*)d_in[9],  (const float*)d_in[15]};
  const float* bias[3]  = {(const float*)d_in[4],  (const float*)d_in[10], (const float*)d_in[16]};
  const float* gamma[3] = {(const float*)d_in[5],  (const float*)d_in[11], (const float*)d_in[17]};
  const float* beta[3]  = {(const float*)d_in[6],  (const float*)d_in[12], (const float*)d_in[18]};
  const float* rmean[3] = {(const float*)d_in[7],  (const float*)d_in[13], (const float*)d_in[19]};
  const float* rvar[3]  = {(const float*)d_in[8],  (const float*)d_in[14], (const float*)d_in[20]};
  const float* Wc1 = (const float*)d_in[21];
  const float* bc1 = (const float*)d_in[22];
  const float* Wc2 = (const float*)d_in[23];
  const float* bc2 = (const float*)d_in[24];
  float* out = (float*)d_out;

  // ---- workspace carve-up (256B aligned) ----
  char* ws = (char*)d_ws;
  size_t off = 0;
  auto carve = [&](size_t bytes) -> char* {
    char* p = ws + off;
    off += (bytes + 255) & ~(size_t)255;
    return p;
  };
  float* T    = (float*)carve((size_t)N_NODES * HID * 4);  // post-GEMM features
  float* AGG  = (float*)carve((size_t)N_NODES * HID * 4);  // scatter accumulator
  float* Hbuf = (float*)carve((size_t)N_NODES * HID * 4);  // post-activation
  float* DINV = (float*)carve((size_t)N_NODES * 4);        // deg, then rsqrt(deg)
  float* POOL = (float*)carve((size_t)N_GRAPHS * HID * 4);
  float* CNT  = (float*)carve((size_t)N_GRAPHS * 4);
  (void)ws_size;

  const int TPB = 256;
  const long nodeWaveThreads = (long)N_NODES * 32;
  const int  nodeWaveBlocks  = (int)((nodeWaveThreads + TPB - 1) / TPB);
  const long edgeWaveThreads = (long)N_EDGES * 32;
  const int  edgeWaveBlocks  = (int)((edgeWaveThreads + TPB - 1) / TPB);
  const int  gemmBlocks      = (N_NODES + 127) / 128;

  // degrees -> dinv
  hipMemsetAsync(DINV, 0, (size_t)N_NODES * 4, stream);
  deg_kernel<<<(N_EDGES + TPB - 1) / TPB, TPB, 0, stream>>>(dst, DINV, N_EDGES);
  dinv_kernel<<<(N_NODES + TPB - 1) / TPB, TPB, 0, stream>>>(DINV, N_NODES);

  for (int l = 0; l < 3; ++l) {
    if (l == 0)
      gemm_in7<<<nodeWaveBlocks, TPB, 0, stream>>>(x, W[0], T, N_NODES);
    else
      gemm128_wmma<<<gemmBlocks, TPB, 0, stream>>>(Hbuf, W[l], T, N_NODES);

    hipMemsetAsync(AGG, 0, (size_t)N_NODES * HID * 4, stream);
    scatter_edges<<<edgeWaveBlocks, TPB, 0, stream>>>(src, dst, DINV, T, AGG, N_EDGES);
    epilogue<<<nodeWaveBlocks, TPB, 0, stream>>>(AGG, T, DINV, bias[l], gamma[l],
                                                 beta[l], rmean[l], rvar[l],
                                                 Hbuf, N_NODES);
  }

  // pooling + head
  hipMemsetAsync(POOL, 0, (size_t)N_GRAPHS * HID * 4, stream);
  hipMemsetAsync(CNT, 0, (size_t)N_GRAPHS * 4, stream);
  graph_count<<<(N_NODES + TPB - 1) / TPB, TPB, 0, stream>>>(batch, CNT, N_NODES);
  pool_sum<<<nodeWaveBlocks, TPB, 0, stream>>>(Hbuf, batch, POOL, N_NODES);

  const long gWaveThreads = (long)N_GRAPHS * 32;
  mlp_head<<<(int)((gWaveThreads + TPB - 1) / TPB), TPB, 0, stream>>>(
      POOL, CNT, Wc1, bc1, Wc2, bc2, out, N_GRAPHS);
}